// DmasifAttentionModule_24584392802394
// MI455X (gfx1250) — compile-verified
//
#include <hip/hip_runtime.h>

#define B_ 16
#define N_ 4096
#define D_ 512
#define NCHUNK 32                 // pool chunks per batch (128 rows each)
#define NEGBIG (-1e9f)

typedef __attribute__((ext_vector_type(2))) float v2f;
typedef __attribute__((ext_vector_type(8))) float v8f;

#if defined(__AMDGCN__)
typedef __attribute__((address_space(1))) int g_int;  // global
typedef __attribute__((address_space(3))) int l_int;  // LDS
#endif

// ---------------------------------------------------------------------------
// Kernel A1: partial u.  Block p handles 32 rows of W:
//   upart[p][d] = sum_{e in [32p,32p+32)} v[e]*W[e,d]      (coalesced in d)
// ---------------------------------------------------------------------------
__global__ __launch_bounds__(512) void prep_uv_part(const float* __restrict__ W,
                                                    const float* __restrict__ v,
                                                    float* __restrict__ upart) {
    const int d  = threadIdx.x;       // 512 threads
    const int p  = blockIdx.x;        // 16 blocks
    const int e0 = p * 32;
    float acc = 0.f;
#pragma unroll 8
    for (int e = e0; e < e0 + 32; ++e) acc = fmaf(v[e], W[e * D_ + d], acc);
    upart[p * D_ + d] = acc;
}

// ---------------------------------------------------------------------------
// Kernel A2: u[d] = sum_p upart[p][d];  c = sum_e bias[e]*v[e] (LDS reduce)
// ---------------------------------------------------------------------------
__global__ __launch_bounds__(512) void prep_uv_reduce(const float* __restrict__ upart,
                                                      const float* __restrict__ bias,
                                                      const float* __restrict__ v,
                                                      float* __restrict__ u,
                                                      float* __restrict__ cptr) {
    const int d = threadIdx.x;  // 512 threads, 1 block
    float acc = 0.f;
#pragma unroll
    for (int p = 0; p < 16; ++p) acc += upart[p * D_ + d];
    u[d] = acc;

    __shared__ float red[512];
    red[d] = bias[d] * v[d];
    __syncthreads();
    for (int off = 256; off > 0; off >>= 1) {
        if (d < off) red[d] += red[d + off];
        __syncthreads();
    }
    if (d == 0) *cptr = red[0];
}

// ---------------------------------------------------------------------------
// Kernel B: scores[b,n] = x[b,n,:]·u + c  (masked -> -1e9)
// WMMA mapping: one wave = 16 consecutive rows (M), K loop over D in steps of 4.
// A layout (f32 16x4): lane L holds M=L%16; lanes>=16 hold K+2/K+3.
// B columns are all u -> identical per-lane K selection.
// D: lane0 holds M=0..7 in acc[0..7], lane16 holds M=8..15.
// ---------------------------------------------------------------------------
#define WMMA_STEP(ACC, KK)                                                     \
    do {                                                                       \
        v2f a_, b_;                                                            \
        a_.x = ap[(KK)];                                                       \
        a_.y = ap[(KK) + 1];                                                   \
        b_.x = ulds[(KK) + koff];                                              \
        b_.y = ulds[(KK) + koff + 1];                                          \
        ACC = __builtin_amdgcn_wmma_f32_16x16x4_f32(false, a_, false, b_,      \
                                                    (short)0, ACC, false,      \
                                                    false);                    \
    } while (0)

__global__ __launch_bounds__(256) void scores_kernel(const float* __restrict__ x,
                                                     const int* __restrict__ mask,
                                                     const float* __restrict__ u,
                                                     const float* __restrict__ cptr,
                                                     float* __restrict__ scores) {
    __shared__ float ulds[D_];
    for (int i = threadIdx.x; i < D_; i += 256) ulds[i] = u[i];
    __syncthreads();

#if defined(__gfx1250__) && __has_builtin(__builtin_amdgcn_wmma_f32_16x16x4_f32)
    const int wave = threadIdx.x >> 5;
    const int lane = threadIdx.x & 31;
    const int tile = blockIdx.x * 8 + wave;  // 4096 tiles of 16 rows
    const int b    = tile >> 8;              // N_/16 = 256 tiles per batch
    const int n0   = (tile & 255) << 4;
    const int m    = lane & 15;
    const int koff = (lane >> 4) << 1;  // 0 or 2

    const float* ap = x + ((size_t)b * N_ + n0 + m) * D_ + koff;

    v8f acc0 = {0.f, 0.f, 0.f, 0.f, 0.f, 0.f, 0.f, 0.f};
    v8f acc1 = {0.f, 0.f, 0.f, 0.f, 0.f, 0.f, 0.f, 0.f};
    v8f acc2 = {0.f, 0.f, 0.f, 0.f, 0.f, 0.f, 0.f, 0.f};
    v8f acc3 = {0.f, 0.f, 0.f, 0.f, 0.f, 0.f, 0.f, 0.f};
#pragma unroll 4
    for (int k0 = 0; k0 < D_; k0 += 16) {  // 4 independent WMMA chains
        WMMA_STEP(acc0, k0 + 0);
        WMMA_STEP(acc1, k0 + 4);
        WMMA_STEP(acc2, k0 + 8);
        WMMA_STEP(acc3, k0 + 12);
    }
    v8f acc = (acc0 + acc1) + (acc2 + acc3);

    const float c = *cptr;
    if (m == 0) {  // lanes 0 and 16 hold the full D column
        const int mbase = (lane >> 4) << 3;  // 0 or 8
        const int base  = b * N_ + n0 + mbase;
#pragma unroll
        for (int r = 0; r < 8; ++r) {
            const float s    = acc[r] + c;
            scores[base + r] = (mask[base + r] == 1) ? s : NEGBIG;
        }
    }
#else
    // VALU fallback (also what the host pass sees): 1 thread = 1 row,
    // only the first 128 threads of each block do work (same grid shape).
    const int t = threadIdx.x;
    if (t < 128) {
        const int g = blockIdx.x * 128 + t;  // over B_*N_ rows
        const int b = g >> 12;
        const int n = g & (N_ - 1);
        const float* row = x + ((size_t)b * N_ + n) * D_;
        float acc = 0.f;
        for (int k = 0; k < D_; ++k) acc = fmaf(row[k], ulds[k], acc);
        scores[g] = (mask[g] == 1) ? (acc + *cptr) : NEGBIG;
    }
#endif
}

// ---------------------------------------------------------------------------
// Kernel C: per-batch softmax over N, in place (scores -> weights)
// ---------------------------------------------------------------------------
__global__ __launch_bounds__(256) void softmax_norm(float* __restrict__ scores) {
    const int b  = blockIdx.x;
    float* s     = scores + b * N_;
    const int t  = threadIdx.x;
    __shared__ float red[256];

    float lmax = -3.4e38f;
    for (int i = t; i < N_; i += 256) lmax = fmaxf(lmax, s[i]);
    red[t] = lmax;
    __syncthreads();
    for (int off = 128; off > 0; off >>= 1) {
        if (t < off) red[t] = fmaxf(red[t], red[t + off]);
        __syncthreads();
    }
    const float m = red[0];
    __syncthreads();

    float lsum = 0.f;
    for (int i = t; i < N_; i += 256) {
        const float e = __expf(s[i] - m);  // masked rows underflow to 0
        s[i] = e;
        lsum += e;
    }
    red[t] = lsum;
    __syncthreads();
    for (int off = 128; off > 0; off >>= 1) {
        if (t < off) red[t] += red[t + off];
        __syncthreads();
    }
    const float inv = 1.f / red[0];
    __syncthreads();
    for (int i = t; i < N_; i += 256) s[i] *= inv;
}

// ---------------------------------------------------------------------------
// Kernel D: partial pooled sums.  Block = (batch, chunk of 128 rows).
// Thread owns 2 consecutive d's -> perfectly coalesced 2KB row loads.
// Rows with weight exactly 0 (masked) are skipped wave-uniformly.
// Weight chunk staged into LDS via CDNA5 async global->LDS + s_wait_asynccnt.
// ---------------------------------------------------------------------------
__global__ __launch_bounds__(256) void pool_partial(const float* __restrict__ x,
                                                    const float* __restrict__ w,
                                                    float* __restrict__ part) {
    const int b     = blockIdx.x / NCHUNK;
    const int chunk = blockIdx.x % NCHUNK;
    const int tid   = threadIdx.x;
    const int rows  = N_ / NCHUNK;  // 128
    __shared__ float wl[N_ / NCHUNK];

    const float* wsrc = w + b * N_ + chunk * rows;
#if defined(__gfx1250__) && __has_builtin(__builtin_amdgcn_global_load_async_to_lds_b32)
    if (tid < rows) {
        __builtin_amdgcn_global_load_async_to_lds_b32(
            (g_int*)(wsrc + tid), (l_int*)(&wl[tid]), 0, 0);
    }
#if __has_builtin(__builtin_amdgcn_s_wait_asynccnt)
    __builtin_amdgcn_s_wait_asynccnt(0);
#else
    asm volatile("s_wait_asynccnt 0" ::: "memory");
#endif
    __syncthreads();
#else
    if (tid < rows) wl[tid] = wsrc[tid];
    __syncthreads();
#endif

    float a0 = 0.f, a1 = 0.f;
    const float* xp = x + ((size_t)b * N_ + chunk * rows) * D_ + tid * 2;
#pragma unroll 4
    for (int r = 0; r < rows; ++r) {
        const float wr = wl[r];  // uniform across the wave
        if (wr != 0.f) {
            const float2 xv = *(const float2*)(xp + (size_t)r * D_);
            a0 = fmaf(wr, xv.x, a0);
            a1 = fmaf(wr, xv.y, a1);
        }
    }
    float* pp = part + ((size_t)(b * NCHUNK + chunk)) * D_ + tid * 2;
    pp[0] = a0;
    pp[1] = a1;
}

// ---------------------------------------------------------------------------
// Kernel E: out[b,d] = sum over NCHUNK partials (deterministic reduce)
// ---------------------------------------------------------------------------
__global__ __launch_bounds__(256) void reduce_out(const float* __restrict__ part,
                                                  float* __restrict__ out) {
    const int g = blockIdx.x * 256 + threadIdx.x;  // B_*D_ = 8192
    const int b = g >> 9;
    const int d = g & (D_ - 1);
    float acc = 0.f;
#pragma unroll
    for (int ch = 0; ch < NCHUNK; ++ch)
        acc += part[((size_t)(b * NCHUNK + ch)) * D_ + d];
    out[g] = acc;
}

// ---------------------------------------------------------------------------
extern "C" void kernel_launch(void* const* d_in, const int* in_sizes, int n_in,
                              void* d_out, int out_size, void* d_ws, size_t ws_size,
                              hipStream_t stream) {
    (void)in_sizes; (void)n_in; (void)out_size; (void)ws_size;

    const float* x    = (const float*)d_in[0];  // [B,N,D]
    const int*   mask = (const int*)d_in[1];    // [B,N]
    const float* W    = (const float*)d_in[2];  // [D,D]
    const float* bias = (const float*)d_in[3];  // [D]
    const float* v    = (const float*)d_in[4];  // [D]
    float*       out  = (float*)d_out;          // [B,D]

    float* wsf    = (float*)d_ws;
    float* u      = wsf;                          // 512
    float* cptr   = wsf + 512;                    // 1
    float* upart  = wsf + 1024;                   // 16*512 = 8192
    float* scores = wsf + 1024 + 8192;            // B*N = 65536 (-> weights)
    float* part   = scores + (size_t)B_ * N_;     // B*NCHUNK*D = 262144

    prep_uv_part<<<16, 512, 0, stream>>>(W, v, upart);
    prep_uv_reduce<<<1, 512, 0, stream>>>(upart, bias, v, u, cptr);
    scores_kernel<<<(B_ * N_) / 128, 256, 0, stream>>>(x, mask, u, cptr, scores);
    softmax_norm<<<B_, 256, 0, stream>>>(scores);
    pool_partial<<<B_ * NCHUNK, 256, 0, stream>>>(x, scores, part);
    reduce_out<<<(B_ * D_) / 256, 256, 0, stream>>>(part, out);
}